// MHA_55370718380664
// MI455X (gfx1250) — compile-verified
//
#include <hip/hip_runtime.h>

// ---------------- problem constants ----------------
#define DIMC   256
#define NPTS   8192
#define BATCH  4
#define HEADS  4
#define DHEAD  64            // DIMC / HEADS
#define KNBR   16
#define RQ     4

#define PTS_WG 4
#define COLS_F (PTS_WG * KNBR)   // 64 feats columns per WG
#define COLS_Q (PTS_WG * RQ)     // 16 query columns per WG
#define LDF    72                // feats/k/v LDS stride: 2*72 %64 = 16 -> bank-safe
#define LDQ    24                // queries/q/agg stride: 2*24 %64 = 48 -> bank-safe
#define LDH1   18                // h1 stride:            2*18 %64 = 36 -> bank-safe
#define NWG    (BATCH * (NPTS / PTS_WG))   // 8192 workgroups

typedef __attribute__((ext_vector_type(2))) float v2f;
typedef __attribute__((ext_vector_type(8))) float v8f;
typedef __attribute__((ext_vector_type(4))) unsigned int v4u;
typedef __attribute__((ext_vector_type(4))) int v4i;
typedef __attribute__((ext_vector_type(8))) int v8i;

#define VZERO8 {0.f,0.f,0.f,0.f,0.f,0.f,0.f,0.f}

#if defined(__AMDGCN__) && __has_builtin(__builtin_amdgcn_tensor_load_to_lds) && \
    __has_builtin(__builtin_amdgcn_s_wait_tensorcnt)
#define USE_TDM 1
#else
#define USE_TDM 0
#endif

// ---------------------------------------------------------------
// WMMA tile kernels (V_WMMA_F32_16X16X4_F32, ISA 7.12.2 layouts):
//  A frag : lane L holds W[(row+L%16)*ldw + c0 + 2*(L/16) + {0,1}]   (float2 global load)
//  B frag : lane L holds P[(c0 + 2*(L/16) + {0,1})*ldb + col+L%16]   (ds_load_2addr)
//  C/D    : lane L, vgpr j holds D[row + j + 8*(L/16)][col + L%16]
// Two (or four) row-tiles share one B-fragment stream -> 2x/4x fewer LDS reads.
__device__ __forceinline__ void wmma_gemm_f32_x2(const float* __restrict__ w0p,
                                                 const float* __restrict__ w1p,
                                                 const float* bptr, int kdim, int ldb,
                                                 v8f& acc0, v8f& acc1) {
#pragma unroll 4
  for (int c0 = 0; c0 < kdim; c0 += 4) {
    v2f a0 = *(const v2f*)(w0p + c0);
    v2f a1 = *(const v2f*)(w1p + c0);
    v2f bb;
    bb.x = bptr[c0 * ldb];
    bb.y = bptr[c0 * ldb + ldb];
    acc0 = __builtin_amdgcn_wmma_f32_16x16x4_f32(false, a0, false, bb, (short)0, acc0, false, false);
    acc1 = __builtin_amdgcn_wmma_f32_16x16x4_f32(false, a1, false, bb, (short)0, acc1, false, false);
  }
}

__device__ __forceinline__ void wmma_gemm_f32_x4(const float* __restrict__ w0p,
                                                 const float* __restrict__ w1p,
                                                 const float* __restrict__ w2p,
                                                 const float* __restrict__ w3p,
                                                 const float* bptr, int kdim, int ldb,
                                                 v8f& acc0, v8f& acc1, v8f& acc2, v8f& acc3) {
#pragma unroll 2
  for (int c0 = 0; c0 < kdim; c0 += 4) {
    v2f a0 = *(const v2f*)(w0p + c0);
    v2f a1 = *(const v2f*)(w1p + c0);
    v2f a2 = *(const v2f*)(w2p + c0);
    v2f a3 = *(const v2f*)(w3p + c0);
    v2f bb;
    bb.x = bptr[c0 * ldb];
    bb.y = bptr[c0 * ldb + ldb];
    acc0 = __builtin_amdgcn_wmma_f32_16x16x4_f32(false, a0, false, bb, (short)0, acc0, false, false);
    acc1 = __builtin_amdgcn_wmma_f32_16x16x4_f32(false, a1, false, bb, (short)0, acc1, false, false);
    acc2 = __builtin_amdgcn_wmma_f32_16x16x4_f32(false, a2, false, bb, (short)0, acc2, false, false);
    acc3 = __builtin_amdgcn_wmma_f32_16x16x4_f32(false, a3, false, bb, (short)0, acc3, false, false);
  }
}

#if USE_TDM
// TDM descriptor group 0 (ISA 8.3): count=1, lds_addr, 57-bit global addr, type=2
__device__ __forceinline__ v4u tdm_group0(unsigned lds, unsigned long long ga) {
  v4u g;
  g[0] = 1u;
  g[1] = lds;
  g[2] = (unsigned)(ga & 0xffffffffull);
  g[3] = (unsigned)((ga >> 32) & 0x01ffffffull) | 0x80000000u;  // type=2 at [127:126]
  return g;
}
// TDM descriptor group 1 (ISA 8.4): 2D tile, data_size=4B, LDS padding for bank-safe stride
__device__ __forceinline__ v8i tdm_group1(unsigned pad_int_code, unsigned pad_amt_code,
                                          unsigned long long tdim0, unsigned tdim1,
                                          unsigned tile0, unsigned tile1,
                                          unsigned long long stride0) {
  v8i g;
  g[0] = (int)((2u << 16) | (1u << 20) | (pad_int_code << 22) | (pad_amt_code << 25));
  g[1] = (int)(((unsigned)(tdim0 & 0xffffull)) << 16);
  g[2] = (int)(((unsigned)(tdim0 >> 16)) & 0xffffu) | (int)((tdim1 & 0xffffu) << 16);
  g[3] = (int)((tdim1 >> 16) & 0xffffu) | (int)(tile0 << 16);
  g[4] = (int)(tile1 & 0xffffu);
  g[5] = (int)(stride0 & 0xffffffffull);
  g[6] = (int)((stride0 >> 32) & 0xffffull);
  g[7] = 0;
  return g;
}
__device__ __forceinline__ void tdm_load_2d(unsigned lds, unsigned long long ga,
                                            unsigned long long tdim0, unsigned tdim1,
                                            unsigned tile0, unsigned tile1,
                                            unsigned long long stride0,
                                            unsigned pad_int_code, unsigned pad_amt_code) {
  v4i gz = {0, 0, 0, 0};
  v4u g0 = tdm_group0(lds, ga);
  v8i g1 = tdm_group1(pad_int_code, pad_amt_code, tdim0, tdim1, tile0, tile1, stride0);
#if defined(__clang_major__) && (__clang_major__ >= 23)
  v8i gz8 = {0, 0, 0, 0, 0, 0, 0, 0};
  __builtin_amdgcn_tensor_load_to_lds(g0, g1, gz, gz, gz8, 0);
#else
  __builtin_amdgcn_tensor_load_to_lds(g0, g1, gz, gz, 0);
#endif
}
#endif  // USE_TDM

// ---------------------------------------------------------------
__global__ void zero_ws_kernel(float* p, int n) {
  int i = blockIdx.x * blockDim.x + threadIdx.x;
  if (i < n) p[i] = 0.f;
}

// ---------------------------------------------------------------
// Stage A: q/k/v projections (WMMA) + per-point attention + residual +
//          W1 GEMM (WMMA) + BN partial sums.  One WG = 4 points of one batch.
__global__ __launch_bounds__(256) void stageA_kernel(
    const float* __restrict__ queries, const float* __restrict__ feats,
    const float* __restrict__ Wq, const float* __restrict__ bq,
    const float* __restrict__ Wk, const float* __restrict__ bk,
    const float* __restrict__ Wv, const float* __restrict__ bv,
    const float* __restrict__ W1, const float* __restrict__ b1,
    float* __restrict__ ws_agg, float* __restrict__ ws_h1,
    float* __restrict__ ws_sum, float* __restrict__ ws_sumsq) {
  extern __shared__ float smem[];
  float* s_feats = smem;                       // 256*LDF
  float* s_k     = s_feats + DIMC * LDF;       // 256*LDF
  float* s_v     = s_k     + DIMC * LDF;       // 256*LDF
  float* s_qin   = s_v     + DIMC * LDF;       // 256*LDQ (input queries panel)
  float* s_qa    = s_qin   + DIMC * LDQ;       // 256*LDQ (projected q, then agg)
  float* s_log   = s_qa    + DIMC * LDQ;       // 1024 logits
  float* s_h1    = s_k;                        // alias: 512*LDH1 <= 256*LDF

  const int tid  = threadIdx.x;
  const int wg   = blockIdx.x;
  const int b    = wg / (NPTS / PTS_WG);
  const int n0   = (wg % (NPTS / PTS_WG)) * PTS_WG;
  const int wave = tid >> 5;
  const int lane = tid & 31;
  const int lm   = lane & 15;   // M / N index within tile
  const int lh   = lane >> 4;   // K-pair selector

  const float* fb = feats + (size_t)b * DIMC * (NPTS * KNBR) + (size_t)n0 * KNBR;
  const float* qb = queries + (size_t)b * DIMC * (NPTS * RQ) + (size_t)n0 * RQ;

#if USE_TDM
  // ---- async Tensor-DMA panel loads: global -> LDS with padded row stride ----
  if (wave == 0) {
    // feats: 256 rows x 64 floats, row stride N*K; LDS stride 72 (pad 8 DW / 64 DW)
    tdm_load_2d((unsigned)(size_t)(void*)s_feats, (unsigned long long)(size_t)fb,
                (unsigned long long)(NPTS * KNBR), DIMC, COLS_F, DIMC,
                (unsigned long long)(NPTS * KNBR), /*pad_int 64DW*/ 5u, /*pad 8DW*/ 7u);
    // queries: 256 rows x 16 floats, row stride N*R; LDS stride 24 (pad 8 DW / 16 DW)
    tdm_load_2d((unsigned)(size_t)(void*)s_qin, (unsigned long long)(size_t)qb,
                (unsigned long long)(NPTS * RQ), DIMC, COLS_Q, DIMC,
                (unsigned long long)(NPTS * RQ), /*pad_int 16DW*/ 3u, /*pad 8DW*/ 7u);
  }
  __builtin_amdgcn_s_wait_tensorcnt(0);
  __syncthreads();
#else
  // ---- fallback: cooperative panel loads (coalesced float4) ----
#pragma unroll
  for (int i = 0; i < 16; ++i) {          // 4096 float4
    int f4 = tid + i * 256;
    int c = f4 >> 4, cc4 = f4 & 15;
    float4 v = *(const float4*)(fb + (size_t)c * (NPTS * KNBR) + cc4 * 4);
    float* d = s_feats + c * LDF + cc4 * 4;
    d[0] = v.x; d[1] = v.y; d[2] = v.z; d[3] = v.w;
  }
#pragma unroll
  for (int i = 0; i < 4; ++i) {           // 1024 float4
    int f4 = tid + i * 256;
    int c = f4 >> 2, cc4 = f4 & 3;
    float4 v = *(const float4*)(qb + (size_t)c * (NPTS * RQ) + cc4 * 4);
    float* d = s_qin + c * LDQ + cc4 * 4;
    d[0] = v.x; d[1] = v.y; d[2] = v.z; d[3] = v.w;
  }
  __syncthreads();
#endif

  // ---- q projection: 256x16; wave -> row-tiles {wave, wave+8}, shared B ----
  {
    v8f acc0 = VZERO8, acc1 = VZERO8;
    wmma_gemm_f32_x2(Wq + (size_t)(wave * 16 + lm) * DIMC + 2 * lh,
                     Wq + (size_t)((wave + 8) * 16 + lm) * DIMC + 2 * lh,
                     s_qin + 2 * lh * LDQ + lm, DIMC, LDQ, acc0, acc1);
#pragma unroll
    for (int j = 0; j < 8; ++j) {
      int r0 = wave * 16 + j + 8 * lh;
      int r1 = (wave + 8) * 16 + j + 8 * lh;
      s_qa[r0 * LDQ + lm] = acc0[j] + bq[r0];
      s_qa[r1 * LDQ + lm] = acc1[j] + bq[r1];
    }
  }
  // ---- k/v projections: 256x64; fixed col-tile per wave, row-tile pairs ----
  {
    const int ct = wave & 3, rb = wave >> 2;
    const float* bp = s_feats + 2 * lh * LDF + ct * 16 + lm;
#pragma unroll
    for (int pp = 0; pp < 4; ++pp) {
      int rt0 = rb + pp * 4, rt1 = rt0 + 2;
      v8f acc0 = VZERO8, acc1 = VZERO8;
      wmma_gemm_f32_x2(Wk + (size_t)(rt0 * 16 + lm) * DIMC + 2 * lh,
                       Wk + (size_t)(rt1 * 16 + lm) * DIMC + 2 * lh,
                       bp, DIMC, LDF, acc0, acc1);
#pragma unroll
      for (int j = 0; j < 8; ++j) {
        int r0 = rt0 * 16 + j + 8 * lh;
        int r1 = rt1 * 16 + j + 8 * lh;
        s_k[r0 * LDF + ct * 16 + lm] = acc0[j] + bk[r0];
        s_k[r1 * LDF + ct * 16 + lm] = acc1[j] + bk[r1];
      }
    }
#pragma unroll
    for (int pp = 0; pp < 4; ++pp) {
      int rt0 = rb + pp * 4, rt1 = rt0 + 2;
      v8f acc0 = VZERO8, acc1 = VZERO8;
      wmma_gemm_f32_x2(Wv + (size_t)(rt0 * 16 + lm) * DIMC + 2 * lh,
                       Wv + (size_t)(rt1 * 16 + lm) * DIMC + 2 * lh,
                       bp, DIMC, LDF, acc0, acc1);
#pragma unroll
      for (int j = 0; j < 8; ++j) {
        int r0 = rt0 * 16 + j + 8 * lh;
        int r1 = rt1 * 16 + j + 8 * lh;
        s_v[r0 * LDF + ct * 16 + lm] = acc0[j] + bv[r0];
        s_v[r1 * LDF + ct * 16 + lm] = acc1[j] + bv[r1];
      }
    }
  }
  __syncthreads();

  // ---- logits[p,h,r,k] = (q_h . k_h) * d^-1/2 ----
#pragma unroll
  for (int j = 0; j < 4; ++j) {
    int id = tid + j * 256;                 // 1024 items
    int kk = id & 15, r = (id >> 4) & 3, h = (id >> 6) & 3, p = id >> 8;
    const float* qp = s_qa + (h * DHEAD) * LDQ + p * RQ + r;
    const float* kp = s_k + (h * DHEAD) * LDF + p * KNBR + kk;
    float acc = 0.f;
#pragma unroll
    for (int dd = 0; dd < DHEAD; ++dd) acc += qp[dd * LDQ] * kp[dd * LDF];
    s_log[id] = acc * 0.125f;               // d = 64 -> 1/sqrt(64)
  }
  __syncthreads();

  // ---- softmax over K=16, one thread per (p,h,r) row ----
  if (tid < 64) {
    float* row = s_log + tid * 16;
    float m = row[0];
#pragma unroll
    for (int i = 1; i < 16; ++i) m = fmaxf(m, row[i]);
    float s = 0.f;
#pragma unroll
    for (int i = 0; i < 16; ++i) { float e = __expf(row[i] - m); row[i] = e; s += e; }
    float inv = 1.f / s;
#pragma unroll
    for (int i = 0; i < 16; ++i) row[i] *= inv;
  }
  __syncthreads();

  // ---- agg = soft @ v + queries (residual); overwrite s_qa; spill to ws ----
#pragma unroll
  for (int j = 0; j < 16; ++j) {
    int id = tid + j * 256;                 // id = c*16 + cc, coalesced ws write
    int cc = id & 15, c = id >> 4;
    int p = cc >> 2, r = cc & 3;
    int h = c >> 6;
    const float* sp = s_log + ((p * HEADS + h) * RQ + r) * 16;
    const float* vp = s_v + c * LDF + p * KNBR;
    float acc = 0.f;
#pragma unroll
    for (int kk = 0; kk < 16; ++kk) acc += sp[kk] * vp[kk];
    acc += s_qin[c * LDQ + cc];             // residual with identity queries
    s_qa[c * LDQ + cc] = acc;
    ws_agg[(size_t)wg * 4096 + id] = acc;
  }
  __syncthreads();

  // ---- h1 = W1 @ agg + b1 : 512x16; wave -> 4 row-tiles, shared B ----
  {
    v8f a0 = VZERO8, a1 = VZERO8, a2 = VZERO8, a3 = VZERO8;
    wmma_gemm_f32_x4(W1 + (size_t)(wave * 16 + lm) * DIMC + 2 * lh,
                     W1 + (size_t)((wave + 8) * 16 + lm) * DIMC + 2 * lh,
                     W1 + (size_t)((wave + 16) * 16 + lm) * DIMC + 2 * lh,
                     W1 + (size_t)((wave + 24) * 16 + lm) * DIMC + 2 * lh,
                     s_qa + 2 * lh * LDQ + lm, DIMC, LDQ, a0, a1, a2, a3);
#pragma unroll
    for (int j = 0; j < 8; ++j) {
      int r0 = wave * 16 + j + 8 * lh;
      int r1 = (wave + 8) * 16 + j + 8 * lh;
      int r2 = (wave + 16) * 16 + j + 8 * lh;
      int r3 = (wave + 24) * 16 + j + 8 * lh;
      s_h1[r0 * LDH1 + lm] = a0[j] + b1[r0];
      s_h1[r1 * LDH1 + lm] = a1[j] + b1[r1];
      s_h1[r2 * LDH1 + lm] = a2[j] + b1[r2];
      s_h1[r3 * LDH1 + lm] = a3[j] + b1[r3];
    }
  }
  __syncthreads();

  // ---- spill h1 (coalesced) + BN partial sums via f32 global atomics ----
#pragma unroll
  for (int i = 0; i < 32; ++i) {
    int idx = tid + i * 256;                // 8192 elems
    ws_h1[(size_t)wg * 8192 + idx] = s_h1[(idx >> 4) * LDH1 + (idx & 15)];
  }
  for (int rr = tid; rr < 2 * DIMC; rr += 256) {
    const float* hp = s_h1 + rr * LDH1;
    float s = 0.f, s2 = 0.f;
#pragma unroll
    for (int cc = 0; cc < 16; ++cc) { float v = hp[cc]; s += v; s2 += v * v; }
    atomicAdd(&ws_sum[rr], s);
    atomicAdd(&ws_sumsq[rr], s2);
  }
}

// ---------------------------------------------------------------
// Stage B: finalize BN statistics (512 channels, count = B*N*R = 131072)
__global__ void stageB_kernel(const float* __restrict__ sum,
                              const float* __restrict__ sumsq,
                              float* __restrict__ mean, float* __restrict__ rstd) {
  int c = blockIdx.x * blockDim.x + threadIdx.x;
  if (c < 2 * DIMC) {
    const float inv = 1.f / (float)(BATCH * NPTS * RQ);
    float m = sum[c] * inv;
    float v = sumsq[c] * inv - m * m;
    mean[c] = m;
    rstd[c] = rsqrtf(v + 1e-5f);
  }
}

// ---------------------------------------------------------------
// Stage C: BN-normalize + ReLU -> W2 GEMM (WMMA) -> +b2 +agg -> ReLU -> out
__global__ __launch_bounds__(256) void stageC_kernel(
    const float* __restrict__ gamma, const float* __restrict__ beta,
    const float* __restrict__ W2, const float* __restrict__ b2,
    const float* __restrict__ ws_agg, const float* __restrict__ ws_h1,
    const float* __restrict__ mean, const float* __restrict__ rstd,
    float* __restrict__ out) {
  extern __shared__ float smem[];
  float* s_h1n = smem;                        // 512*LDH1
  const int tid  = threadIdx.x;
  const int wg   = blockIdx.x;
  const int b    = wg / (NPTS / PTS_WG);
  const int n0   = (wg % (NPTS / PTS_WG)) * PTS_WG;
  const int wave = tid >> 5;
  const int lane = tid & 31;
  const int lm   = lane & 15;
  const int lh   = lane >> 4;

#pragma unroll
  for (int i = 0; i < 32; ++i) {
    int idx = tid + i * 256;
    int c = idx >> 4, cc = idx & 15;
    float v = ws_h1[(size_t)wg * 8192 + idx];
    v = (v - mean[c]) * rstd[c] * gamma[c] + beta[c];
    s_h1n[c * LDH1 + cc] = fmaxf(v, 0.f);
  }
  __syncthreads();

  // ---- h2 = W2 @ h1n : 256x16; wave -> row-tiles {wave, wave+8}, shared B ----
  {
    v8f acc0 = VZERO8, acc1 = VZERO8;
    wmma_gemm_f32_x2(W2 + (size_t)(wave * 16 + lm) * (2 * DIMC) + 2 * lh,
                     W2 + (size_t)((wave + 8) * 16 + lm) * (2 * DIMC) + 2 * lh,
                     s_h1n + 2 * lh * LDH1 + lm, 2 * DIMC, LDH1, acc0, acc1);
#pragma unroll
    for (int j = 0; j < 8; ++j) {
      int r0 = wave * 16 + j + 8 * lh;
      int r1 = (wave + 8) * 16 + j + 8 * lh;
      float v0 = acc0[j] + b2[r0] + ws_agg[(size_t)wg * 4096 + r0 * 16 + lm];
      float v1 = acc1[j] + b2[r1] + ws_agg[(size_t)wg * 4096 + r1 * 16 + lm];
      v0 = fmaxf(v0, 0.f);
      v1 = fmaxf(v1, 0.f);
      // out layout [B, DIM, N, R]; local col lm -> (n = n0 + lm/4, r = lm%4)
      out[(size_t)(b * DIMC + r0) * (NPTS * RQ) + (size_t)n0 * RQ + lm] = v0;
      out[(size_t)(b * DIMC + r1) * (NPTS * RQ) + (size_t)n0 * RQ + lm] = v1;
    }
  }
}

// ---------------------------------------------------------------
extern "C" void kernel_launch(void* const* d_in, const int* in_sizes, int n_in,
                              void* d_out, int out_size, void* d_ws, size_t ws_size,
                              hipStream_t stream) {
  const float* queries = (const float*)d_in[0];
  const float* feats   = (const float*)d_in[1];
  const float* Wq = (const float*)d_in[2];  const float* bq = (const float*)d_in[3];
  const float* Wk = (const float*)d_in[4];  const float* bk = (const float*)d_in[5];
  const float* Wv = (const float*)d_in[6];  const float* bv = (const float*)d_in[7];
  const float* W1 = (const float*)d_in[8];  const float* b1 = (const float*)d_in[9];
  const float* gamma = (const float*)d_in[10];
  const float* beta  = (const float*)d_in[11];
  const float* W2 = (const float*)d_in[12]; const float* b2 = (const float*)d_in[13];

  float* ws      = (float*)d_ws;
  float* ws_sum   = ws;               // 512
  float* ws_sumsq = ws + 512;         // 512
  float* ws_mean  = ws + 1024;        // 512
  float* ws_rstd  = ws + 1536;        // 512
  float* ws_agg   = ws + 2048;                          // NWG*4096 floats (128 MB)
  float* ws_h1    = ws_agg + (size_t)NWG * 4096;        // NWG*8192 floats (256 MB)

  zero_ws_kernel<<<4, 256, 0, stream>>>(ws_sum, 1024);

  size_t shA = (size_t)(3 * DIMC * LDF + 2 * DIMC * LDQ + 1024) * sizeof(float);
  stageA_kernel<<<NWG, 256, shA, stream>>>(queries, feats, Wq, bq, Wk, bk, Wv, bv,
                                           W1, b1, ws_agg, ws_h1, ws_sum, ws_sumsq);

  stageB_kernel<<<2, 256, 0, stream>>>(ws_sum, ws_sumsq, ws_mean, ws_rstd);

  size_t shC = (size_t)(2 * DIMC * LDH1) * sizeof(float);
  stageC_kernel<<<NWG, 256, shC, stream>>>(gamma, beta, W2, b2, ws_agg, ws_h1,
                                           ws_mean, ws_rstd, (float*)d_out);
}